// Dense3DSpatialTransformer_8014408974511
// MI455X (gfx1250) — compile-verified
//
#include <hip/hip_runtime.h>

#ifndef __has_builtin
#define __has_builtin(x) 0
#endif

// Problem constants (B,H,W,D,C = 4,160,160,160,1)
constexpr int Bc  = 4;
constexpr int Hc  = 160;
constexpr int Wc  = 160;
constexpr int Dc  = 160;
constexpr int HWD = Hc * Wc * Dc;          // 4,096,000

// Block covers one (b,h,w) row along D: 160 threads = 5 waves (wave32).
// Grid (W,H,B) -> b/h/w are scalar (SALU), no per-thread div/mod.
constexpr int BLOCK = Dc;                  // 160

// Probe: async global->LDS path (gfx1250 ASYNCcnt instructions)
#if defined(__gfx1250__) && \
    __has_builtin(__builtin_amdgcn_global_load_async_to_lds_b128) && \
    __has_builtin(__builtin_amdgcn_s_wait_asynccnt)
#define USE_ASYNC_LDS 1
#else
#define USE_ASYNC_LDS 0
#endif

#if USE_ASYNC_LDS
// Exact parameter types per hipcc diagnostic: int vector_size(16), AS(1) src.
typedef int v4i __attribute__((vector_size(16)));
typedef __attribute__((address_space(1))) v4i g_v4i;   // global
typedef __attribute__((address_space(3))) v4i l_v4i;   // LDS
#endif

__global__ __launch_bounds__(BLOCK)
void dense3d_warp_kernel(const float* __restrict__ image,
                         const float* __restrict__ flow,
                         float* __restrict__ out)
{
    // Flow slab for this row: 160 voxels * 3 floats = 480 floats = 1920 B,
    // contiguous and 16B-aligned (1920 * blockRow). Staged via B128 async.
    __shared__ float sflow[Dc * 3];

    const int d = threadIdx.x;       // depth index (innermost, stride 1)
    const int w = blockIdx.x;        // width
    const int h = blockIdx.y;        // height
    const int b = blockIdx.z;        // batch

    // All scalar math (SGPR): row base in voxels.
    const int row_base = ((b * Hc + h) * Wc + w) * Dc;   // vox0
    const int vox = row_base + d;

    // Flow ~ N(0,1): gather targets cluster around the identity voxel.
    // Warm that cacheline at WGP scope (all cache levels) before the
    // data-dependent gathers issue. Locality 3 -> near-cache prefetch.
    __builtin_prefetch(&image[vox], 0, 3);

    const float* fsrc = flow + (long long)row_base * 3;

#if USE_ASYNC_LDS
    if (d < (Dc * 3) / 4) {  // 120 lanes move 16 B each
        __builtin_amdgcn_global_load_async_to_lds_b128(
            (g_v4i*)(fsrc + d * 4), (l_v4i*)(&sflow[d * 4]),
            /*offset=*/0, /*cpol=*/0);
    }
    __builtin_amdgcn_s_wait_asynccnt(0);
    __syncthreads();
#else
    if (d < (Dc * 3) / 4) {
        ((float4*)sflow)[d] = ((const float4*)fsrc)[d];
    }
    __syncthreads();
#endif

    // Reference: dx = flow[...,1], dy = flow[...,0], dz = flow[...,2]
    // Stride-3 dword reads: 3d mod 64 distinct within a wave -> conflict-free.
    const float dy = sflow[d * 3 + 0];
    const float dx = sflow[d * 3 + 1];
    const float dz = sflow[d * 3 + 2];

    const float x = dx + (float)w;   // x indexes width
    const float y = dy + (float)h;   // y indexes height
    const float z = dz + (float)d;   // z indexes depth

    const int x0u = (int)floorf(x);
    const int y0u = (int)floorf(y);
    const int z0u = (int)floorf(z);

    // Clip order matches reference: x1 from UNclipped x0+1; weights use clipped x1.
    const int x1 = min(max(x0u + 1, 0), Wc - 1);
    const int y1 = min(max(y0u + 1, 0), Hc - 1);
    const int z1 = min(max(z0u + 1, 0), Dc - 1);
    const int x0 = min(max(x0u, 0), Wc - 1);
    const int y0 = min(max(y0u, 0), Hc - 1);
    const int z0 = min(max(z0u, 0), Dc - 1);

    const float ddx = (float)x1 - x;
    const float ddy = (float)y1 - y;
    const float ddz = (float)z1 - z;
    const float ex = 1.0f - ddx;
    const float ey = 1.0f - ddy;
    const float ez = 1.0f - ddz;

    // b is wave-uniform -> SGPR base pointer; gathers use saddr + 32-bit voffset.
    const float* imb = image + b * HWD;
    const int ry0x0 = (y0 * Wc + x0) * Dc;
    const int ry1x0 = (y1 * Wc + x0) * Dc;
    const int ry0x1 = (y0 * Wc + x1) * Dc;
    const int ry1x1 = (y1 * Wc + x1) * Dc;

    // 8 gathers; z0/z1 adjacent -> same cacheline almost always (L0/L2 hits).
    const float va = imb[ry0x0 + z0];   // (y0,x0,z0)  w = ddz*ddx*ddy
    const float vb = imb[ry1x0 + z0];   // (y1,x0,z0)  w = ddz*ddx*ey
    const float vc = imb[ry0x1 + z0];   // (y0,x1,z0)  w = ddz*ex*ddy
    const float vd = imb[ry1x1 + z0];   // (y1,x1,z0)  w = ddz*ex*ey
    const float ve = imb[ry0x0 + z1];   // (y0,x0,z1)  w = ez*ddx*ddy
    const float vf = imb[ry1x0 + z1];   // (y1,x0,z1)  w = ez*ddx*ey
    const float vg = imb[ry0x1 + z1];   // (y0,x1,z1)  w = ez*ex*ddy
    const float vh = imb[ry1x1 + z1];   // (y1,x1,z1)  w = ez*ex*ey

    // Factored trilinear blend (FMA tree).
    const float lz0 = fmaf(ddx, fmaf(ddy, va, ey * vb),
                           ex * fmaf(ddy, vc, ey * vd));
    const float lz1 = fmaf(ddx, fmaf(ddy, ve, ey * vf),
                           ex * fmaf(ddy, vg, ey * vh));
    const float result = fmaf(ddz, lz0, ez * lz1);

    // Output is write-once / never re-read: non-temporal store keeps the
    // 65.5 MB of dead output lines from evicting the L2-resident image.
    __builtin_nontemporal_store(result, &out[vox]);
}

extern "C" void kernel_launch(void* const* d_in, const int* in_sizes, int n_in,
                              void* d_out, int out_size, void* d_ws, size_t ws_size,
                              hipStream_t stream)
{
    const float* image = (const float*)d_in[0];  // [B,H,W,D,1] fp32
    const float* flow  = (const float*)d_in[1];  // [B,H,W,D,3] fp32
    float* out = (float*)d_out;                  // [B,H,W,D,1] fp32

    dim3 grid(Wc, Hc, Bc);                       // 160 x 160 x 4 = 102,400 blocks
    dense3d_warp_kernel<<<grid, BLOCK, 0, stream>>>(image, flow, out);
}